// Tianfu_MultiviewReflectionMappingShader_85237920957132
// MI455X (gfx1250) — compile-verified
//
#include <hip/hip_runtime.h>
#include <hip/hip_bf16.h>
#include <math.h>

typedef __attribute__((ext_vector_type(2))) float v2f;
typedef __attribute__((ext_vector_type(8))) float v8f;

#define EPS_F 1e-6f

__device__ __forceinline__ float lane_gather(float v, int src) {
  return __int_as_float(__builtin_amdgcn_ds_bpermute(src << 2, __float_as_int(v)));
}

__device__ __forceinline__ float sel3(int k, float s0, float s1, float s2) {
  float r = (k == 0) ? s0 : 0.0f;
  r = (k == 1) ? s1 : r;
  r = (k == 2) ? s2 : r;
  return r;
}

// Row-vector transform out = [x y z w] * A^T-layout via V_WMMA_F32_16X16X4_F32.
// A (16x4): rows 0..2 = transform (A[j][i] = M[i][j], A[j][3] = T[j]), rest 0.
// B (4x16): 16 pixel columns (x,y,z,w). Two WMMAs cover the 32 wave pixels.
// A VGPR layout (ISA 7.12.2): a0 = K0 | K2 , a1 = K1 | K3 (lanes 0-15 | 16-31).
// B VGPR layout (rows striped across lanes): b0 = rowK0 | rowK2, b1 = rowK1 | rowK3.
__device__ __forceinline__ void wmma_rowvec_xform(
    float a0, float a1,
    float inx, float iny, float inz, float inw,
    float& ox, float& oy, float& oz)
{
  const int lane = (int)(threadIdx.x & 31u);
  const int l15 = lane & 15;
  const bool lo = (lane < 16);
  v2f A; A.x = a0; A.y = a1;
  v8f cz = {0.0f, 0.0f, 0.0f, 0.0f, 0.0f, 0.0f, 0.0f, 0.0f};

  // group 0: wave pixels 0..15 as columns
  float bx0 = lane_gather(inx, l15);
  float by0 = lane_gather(iny, l15);
  float bz0 = lane_gather(inz, l15);
  float bw0 = lane_gather(inw, l15);
  v2f B0; B0.x = lo ? bx0 : bz0; B0.y = lo ? by0 : bw0;
  v8f D0 = __builtin_amdgcn_wmma_f32_16x16x4_f32(false, A, false, B0, (short)0, cz, false, false);

  // group 1: wave pixels 16..31 as columns
  const int hi = l15 + 16;
  float bx1 = lane_gather(inx, hi);
  float by1 = lane_gather(iny, hi);
  float bz1 = lane_gather(inz, hi);
  float bw1 = lane_gather(inw, hi);
  v2f B1; B1.x = lo ? bx1 : bz1; B1.y = lo ? by1 : bw1;
  v8f D1 = __builtin_amdgcn_wmma_f32_16x16x4_f32(false, A, false, B1, (short)0, cz, false, false);

  // D rows 0..2 live in vgprs 0..2, lanes 0..15 = pixel columns of that group.
  float x1 = lane_gather(D1[0], l15);
  float y1 = lane_gather(D1[1], l15);
  float z1 = lane_gather(D1[2], l15);
  ox = lo ? D0[0] : x1;
  oy = lo ? D0[1] : y1;
  oz = lo ? D0[2] : z1;
}

__global__ __launch_bounds__(256) void Tianfu_MultiviewReflectionMappingShader_85237920957132_kernel(
    const float* __restrict__ verts,
    const float* __restrict__ vnorm,
    const float* __restrict__ campos,
    const float* __restrict__ bary,
    const float* __restrict__ tex,
    const float* __restrict__ Rm,
    const float* __restrict__ tvec,
    const int*  __restrict__ faces,
    const int*  __restrict__ p2f,
    float* __restrict__ out,
    int total)
{
  constexpr int H = 1024, W = 1024, Nv = 4, Nt = 2, C = 3, Hi = 1024, Wi = 1024;
  constexpr int HW = H * W;  // 1<<20

  const int p = blockIdx.x * blockDim.x + threadIdx.x;
  if (p >= total) return;  // total is a multiple of 256 -> EXEC stays all-ones
  const int nv = p >> 20;
  const int hw = p & (HW - 1);

  // ---- uniform transform data (scalar loads) ----
  float r00 = Rm[0], r01 = Rm[1], r02 = Rm[2];
  float r10 = Rm[3], r11 = Rm[4], r12 = Rm[5];
  float r20 = Rm[6], r21 = Rm[7], r22 = Rm[8];
  float t0 = tvec[0], t1 = tvec[1], t2 = tvec[2];

  // inv(R) via adjugate
  float det = r00 * (r11 * r22 - r12 * r21)
            + r01 * (r12 * r20 - r10 * r22)
            + r02 * (r10 * r21 - r11 * r20);
  float id = 1.0f / det;
  float i00 = (r11 * r22 - r12 * r21) * id;
  float i01 = (r02 * r21 - r01 * r22) * id;
  float i02 = (r01 * r12 - r02 * r11) * id;
  float i10 = (r12 * r20 - r10 * r22) * id;
  float i11 = (r00 * r22 - r02 * r20) * id;
  float i12 = (r02 * r10 - r00 * r12) * id;
  float i20 = (r10 * r21 - r11 * r20) * id;
  float i21 = (r01 * r20 - r00 * r21) * id;
  float i22 = (r00 * r11 - r01 * r10) * id;

  // ---- gather + barycentric interpolation ----
  int f = p2f[p];
  bool valid = (f >= 0);
  int idx = valid ? f : 0;
  float vmask = valid ? 1.0f : 0.0f;

  int v0i = faces[idx * 3 + 0];
  int v1i = faces[idx * 3 + 1];
  int v2i = faces[idx * 3 + 2];
  float b0 = bary[(size_t)p * 3 + 0];
  float b1 = bary[(size_t)p * 3 + 1];
  float b2 = bary[(size_t)p * 3 + 2];

  float pvx = (b0 * verts[v0i * 3 + 0] + b1 * verts[v1i * 3 + 0] + b2 * verts[v2i * 3 + 0]) * vmask;
  float pvy = (b0 * verts[v0i * 3 + 1] + b1 * verts[v1i * 3 + 1] + b2 * verts[v2i * 3 + 1]) * vmask;
  float pvz = (b0 * verts[v0i * 3 + 2] + b1 * verts[v1i * 3 + 2] + b2 * verts[v2i * 3 + 2]) * vmask;

  float pnx = (b0 * vnorm[v0i * 3 + 0] + b1 * vnorm[v1i * 3 + 0] + b2 * vnorm[v2i * 3 + 0]) * vmask;
  float pny = (b0 * vnorm[v0i * 3 + 1] + b1 * vnorm[v1i * 3 + 1] + b2 * vnorm[v2i * 3 + 1]) * vmask;
  float pnz = (b0 * vnorm[v0i * 3 + 2] + b1 * vnorm[v1i * 3 + 2] + b2 * vnorm[v2i * 3 + 2]) * vmask;

  float nl = sqrtf(pnx * pnx + pny * pny + pnz * pnz);
  float inl = 1.0f / fmaxf(nl, EPS_F);
  pnx *= inl; pny *= inl; pnz *= inl;

  float cx = campos[nv * 3 + 0] - pvx;
  float cy = campos[nv * 3 + 1] - pvy;
  float czv = campos[nv * 3 + 2] - pvz;
  float vl = sqrtf(cx * cx + cy * cy + czv * czv);
  float ivl = 1.0f / fmaxf(vl, EPS_F);
  float vdx = cx * ivl, vdy = cy * ivl, vdz = czv * ivl;

  float cosv = pnx * vdx + pny * vdy + pnz * vdz;
  float rfx = 2.0f * cosv * pnx - vdx;
  float rfy = 2.0f * cosv * pny - vdy;
  float rfz = 2.0f * cosv * pnz - vdz;

  // ---- two 3x3 row-vector transforms on tensor cores ----
  const int lane = (int)(threadIdx.x & 31u);
  const int l15 = lane & 15;
  const bool lo = (lane < 16);

  // transform 1: screen_verts = pv @ R + t  (homogeneous w = 1)
  float a0A = lo ? sel3(l15, r00, r01, r02) : sel3(l15, r20, r21, r22);
  float a1A = lo ? sel3(l15, r10, r11, r12) : sel3(l15, t0, t1, t2);
  float svx, svy, svz;
  wmma_rowvec_xform(a0A, a1A, pvx, pvy, pvz, 1.0f, svx, svy, svz);

  // transform 2: reflect_screen = reflect @ inv(R)  (w = 0, no translation)
  float a0B = lo ? sel3(l15, i00, i01, i02) : sel3(l15, i20, i21, i22);
  float a1B = lo ? sel3(l15, i10, i11, i12) : 0.0f;
  float rsx, rsy, rsz;
  wmma_rowvec_xform(a0B, a1B, rfx, rfy, rfz, 0.0f, rsx, rsy, rsz);

  // ---- ray intersection -> uv ----
  float zc = fminf(rsz, -1e-10f);
  float ray = -svz / zc;
  float isx = svx + ray * rsx;
  float isy = svy + ray * rsy;
  float u = isx;          //  * (2/SCREEN_U) == 1
  float vv = -isy;        //  * (-2/SCREEN_V) == -1

  // ---- bilinear sample, fused over both textures ----
  float xg = (u + 1.0f) * (Wi * 0.5f) - 0.5f;
  float yg = (vv + 1.0f) * (Hi * 0.5f) - 0.5f;
  float x0f = floorf(xg), y0f = floorf(yg);
  float wx1 = xg - x0f, wy1 = yg - y0f;
  float wx0 = 1.0f - wx1, wy0 = 1.0f - wy1;

  int x0 = (int)fminf(fmaxf(x0f, -2.0f), (float)(Wi + 1));
  int y0 = (int)fminf(fmaxf(y0f, -2.0f), (float)(Hi + 1));
  int x1 = x0 + 1, y1 = y0 + 1;

  bool vx0 = (x0 >= 0) & (x0 < Wi);
  bool vx1 = (x1 >= 0) & (x1 < Wi);
  bool vy0 = (y0 >= 0) & (y0 < Hi);
  bool vy1 = (y1 >= 0) & (y1 < Hi);
  int x0c = min(max(x0, 0), Wi - 1);
  int x1c = min(max(x1, 0), Wi - 1);
  int y0c = min(max(y0, 0), Hi - 1);
  int y1c = min(max(y1, 0), Hi - 1);

  float w00 = wx0 * wy0 * ((vx0 & vy0) ? 1.0f : 0.0f);
  float w10 = wx1 * wy0 * ((vx1 & vy0) ? 1.0f : 0.0f);
  float w01 = wx0 * wy1 * ((vx0 & vy1) ? 1.0f : 0.0f);
  float w11 = wx1 * wy1 * ((vx1 & vy1) ? 1.0f : 0.0f);

  int o00 = y0c * Wi + x0c;
  int o10 = y0c * Wi + x1c;
  int o01 = y1c * Wi + x0c;
  int o11 = y1c * Wi + x1c;

#pragma unroll
  for (int tm = 0; tm < Nt; ++tm) {
#pragma unroll
    for (int c = 0; c < C; ++c) {
      const float* plane = tex + (size_t)((tm * C + c) * Hi) * Wi;
      float acc = w00 * plane[o00] + w10 * plane[o10]
                + w01 * plane[o01] + w11 * plane[o11];
      out[(((size_t)tm * Nv + nv) * HW + hw) * C + c] = acc;
    }
  }
}

extern "C" void kernel_launch(void* const* d_in, const int* in_sizes, int n_in,
                              void* d_out, int out_size, void* d_ws, size_t ws_size,
                              hipStream_t stream) {
  const float* verts  = (const float*)d_in[0];
  const float* vnorm  = (const float*)d_in[1];
  const float* campos = (const float*)d_in[2];
  const float* bary   = (const float*)d_in[3];
  const float* tex    = (const float*)d_in[4];
  const float* Rm     = (const float*)d_in[5];
  const float* tvec   = (const float*)d_in[6];
  const int*   faces  = (const int*)d_in[7];
  const int*   p2f    = (const int*)d_in[8];
  float* out = (float*)d_out;

  const int total = in_sizes[8];  // Nv*H*W (4,194,304) — multiple of 256
  dim3 block(256);
  dim3 grid((unsigned)((total + 255) / 256));
  hipLaunchKernelGGL(Tianfu_MultiviewReflectionMappingShader_85237920957132_kernel,
                     grid, block, 0, stream,
                     verts, vnorm, campos, bary, tex, Rm, tvec, faces, p2f, out, total);
}